// PatchEmbed_80298708566629
// MI455X (gfx1250) — compile-verified
//
#include <hip/hip_runtime.h>
#include <hip/hip_bf16.h>

// PatchEmbed permute: x (N=16, C=64, H=256, W=256) f32, patch P=4
//   out[n][jb*Hp+ib][c*P*P + r*P + s] = x[n][c][ib*P+r][jb*P+s]
// plus ori_shape int32[4] = {N,C,H,W} appended after the float output.
//
// Pure data movement: 256 MiB in + 256 MiB out => ~22 us at 23.3 TB/s HBM.
// Strategy: float4 (b128) on both sides; one 256-thread block per patch so
// stores are perfectly contiguous (4 KiB/block); grid ordered jb-fastest so
// the 8 blocks sharing each 128B input cacheline are launched back-to-back.

constexpr int N_ = 16;
constexpr int C_ = 64;
constexpr int H_ = 256;
constexpr int W_ = 256;
constexpr int P_ = 4;
constexpr int Hp_ = H_ / P_;                  // 64
constexpr int Wp_ = W_ / P_;                  // 64
constexpr int PATCHES_ = Hp_ * Wp_;           // 4096
constexpr int PATCH_LEN_ = C_ * P_ * P_;      // 1024 floats per patch
constexpr int VEC_PER_PATCH_ = PATCH_LEN_ / 4;// 256 float4 per patch
constexpr unsigned long long OUT_ELEMS_ =
    (unsigned long long)N_ * PATCHES_ * PATCH_LEN_;   // 67,108,864 floats

__global__ __launch_bounds__(256) void
patch_embed_permute_kernel(const float* __restrict__ x, float* __restrict__ out) {
    // blockIdx.x = ((n*Hp + ib)*Wp + jb), jb fastest for input-line sharing.
    const int b  = blockIdx.x;
    const int jb = b & (Wp_ - 1);
    const int ib = (b >> 6) & (Hp_ - 1);
    const int n  = b >> 12;

    const int t = threadIdx.x;   // 0..255, one float4 of this patch
    const int c = t >> 2;        // channel 0..63
    const int r = t & 3;         // patch row 0..3

    // Input: 16B chunk at row (ib*P + r) of channel c, columns [jb*P, jb*P+4)
    const float* src_f =
        x + ((((n * C_ + c) * H_) + (ib * P_ + r)) * W_ + jb * P_);
    const float4 v = *reinterpret_cast<const float4*>(src_f);

    // Prefetch the row 4 lines down: exact data for the block 64 ahead
    // (same jb, next ib) -> warms GL2 along the launch order.
    __builtin_prefetch(src_f + P_ * W_, 0, 1);

    // Output: patch index has jb outer, ib inner; writes fully contiguous.
    const int patch = jb * Hp_ + ib;
    float4* dst = reinterpret_cast<float4*>(out) +
                  ((unsigned)(n * PATCHES_ + patch)) * VEC_PER_PATCH_ + t;
    *dst = v;

    // Tuple output #2: ori_shape int32[4] appended after the floats.
    if (b == 0 && t < 4) {
        const int dims[4] = {N_, C_, H_, W_};
        reinterpret_cast<int*>(out + OUT_ELEMS_)[t] = dims[t];
    }
}

extern "C" void kernel_launch(void* const* d_in, const int* in_sizes, int n_in,
                              void* d_out, int out_size, void* d_ws, size_t ws_size,
                              hipStream_t stream) {
    (void)in_sizes; (void)n_in; (void)d_ws; (void)ws_size; (void)out_size;
    const float* x = (const float*)d_in[0];
    float* out = (float*)d_out;

    const int blocks = N_ * Hp_ * Wp_;   // 65536 blocks, one patch each
    patch_embed_permute_kernel<<<blocks, 256, 0, stream>>>(x, out);
}